// GCN_cat_47218870452451
// MI455X (gfx1250) — compile-verified
//
#include <hip/hip_runtime.h>
#include <stdint.h>

// GCN_cat on MI455X (gfx1250, wave32). All matrix math via V_WMMA_F32_16X16X4_F32
// (exact fp32 -> matches the fp32 reference's threshold + accumulation semantics).

#define N_NODES 8192
#define NFEAT   500          // divisible by 4 -> clean K loop for 16x16x4 WMMA
#define NHID    256
#define NT      (N_NODES / 16)   // 512 16-row tiles
#define MASK_LD (N_NODES / 16)   // u16 words per mask row

typedef float v2f __attribute__((ext_vector_type(2)));
typedef float v8f __attribute__((ext_vector_type(8)));

static __device__ __forceinline__ v8f wmma_f32(v2f a, v2f b, v8f c) {
  // (neg_a, A, neg_b, B, c_mod, C, reuse_a, reuse_b)
  return __builtin_amdgcn_wmma_f32_16x16x4_f32(false, a, false, b, (short)0, c,
                                               false, false);
}

// ---------------------------------------------------------------- utilities
__global__ __launch_bounds__(256) void k_zero(unsigned* __restrict__ p, int n) {
  int i = blockIdx.x * 256 + threadIdx.x;
  if (i < n) p[i] = 0u;
}

// sq[i] = sum_k X[i,k]^2 ; one wave per row
__global__ __launch_bounds__(256) void k_sq(const float* __restrict__ X,
                                            float* __restrict__ sq) {
  int lane = threadIdx.x & 31, wid = threadIdx.x >> 5;
  int row = blockIdx.x * 8 + wid;
  const float* xr = X + (size_t)row * NFEAT;
  float s = 0.f;
  for (int k = lane; k < NFEAT; k += 32) { float v = xr[k]; s += v * v; }
  for (int off = 16; off; off >>= 1) s += __shfl_xor(s, off, 32);
  if (lane == 0) sq[row] = s;
}

// --------------------------------------------- pass 1: max of pairwise d2
// One wave per 16x16 tile of G = X X^T, upper triangle only (d2 symmetric).
__global__ __launch_bounds__(256) void k_gram_max(const float* __restrict__ X,
                                                  const float* __restrict__ sq,
                                                  float* __restrict__ blkmax) {
  __shared__ float smax[8];
  int lane = threadIdx.x & 31, wid = threadIdx.x >> 5;
  int tn = blockIdx.x * 8 + wid, tm = blockIdx.y;
  float mx = 0.f;  // d2 clamped >= 0 anyway
  if (tn >= tm) {
    int q = lane & 15, half = lane >> 4, ka = half * 2;
    int m0 = tm * 16, n0 = tn * 16;
    const float* ar = X + (size_t)(m0 + q) * NFEAT + ka;  // A: rows m0..m0+15
    const float* br = X + (size_t)(n0 + q) * NFEAT + ka;  // B = X^T tile
    v8f acc = {0.f, 0.f, 0.f, 0.f, 0.f, 0.f, 0.f, 0.f};
    for (int k = 0; k < NFEAT; k += 4) {
      v2f a = {ar[k], ar[k + 1]};
      v2f b = {br[k], br[k + 1]};
      acc = wmma_f32(a, b, acc);
    }
    float sqn = sq[n0 + q];
    int mb = m0 + half * 8;
    for (int v = 0; v < 8; ++v)
      mx = fmaxf(mx, sq[mb + v] + sqn - 2.f * acc[v]);
  }
  for (int off = 16; off; off >>= 1) mx = fmaxf(mx, __shfl_xor(mx, off, 32));
  if (lane == 0) smax[wid] = mx;
  __syncthreads();
  if (threadIdx.x == 0) {
    float m = smax[0];
    for (int i = 1; i < 8; ++i) m = fmaxf(m, smax[i]);
    blkmax[blockIdx.y * gridDim.x + blockIdx.x] = m;
  }
}

__global__ __launch_bounds__(256) void k_maxred(const float* __restrict__ blkmax,
                                                int n, float* __restrict__ tmax) {
  __shared__ float sm[8];
  float m = 0.f;
  for (int i = threadIdx.x; i < n; i += 256) m = fmaxf(m, blkmax[i]);
  for (int off = 16; off; off >>= 1) m = fmaxf(m, __shfl_xor(m, off, 32));
  int lane = threadIdx.x & 31, wid = threadIdx.x >> 5;
  if (lane == 0) sm[wid] = m;
  __syncthreads();
  if (threadIdx.x == 0) {
    for (int i = 1; i < 8; ++i) m = fmaxf(m, sm[i]);
    *tmax = m;
  }
}

// ------------------------------ pass 2: recompute d2, emit mask bits + deg
// mask16[r*MASK_LD + w] bit j <=> edge r -> (w*16+j), only r < c stored.
__global__ __launch_bounds__(256) void k_gram_mask(
    const float* __restrict__ X, const float* __restrict__ sq,
    const float* __restrict__ tmax, unsigned short* __restrict__ mask16,
    unsigned* __restrict__ deg) {
  int lane = threadIdx.x & 31, wid = threadIdx.x >> 5;
  int tn = blockIdx.x * 8 + wid, tm = blockIdx.y;
  if (tn < tm) return;  // wave-uniform
  float t = 0.5f * tmax[0];
  int q = lane & 15, half = lane >> 4, ka = half * 2;
  int m0 = tm * 16, n0 = tn * 16;
  const float* ar = X + (size_t)(m0 + q) * NFEAT + ka;
  const float* br = X + (size_t)(n0 + q) * NFEAT + ka;
  v8f acc = {0.f, 0.f, 0.f, 0.f, 0.f, 0.f, 0.f, 0.f};
  for (int k = 0; k < NFEAT; k += 4) {
    v2f a = {ar[k], ar[k + 1]};
    v2f b = {br[k], br[k + 1]};
    acc = wmma_f32(a, b, acc);
  }
  float sqn = sq[n0 + q];
  int c = n0 + q;
  unsigned cnt = 0;
  for (int v = 0; v < 8; ++v) {
    int r = m0 + half * 8 + v;
    float d2 = fmaxf(sq[r] + sqn - 2.f * acc[v], 0.f);
    bool pred = (r < c) && (d2 < t);
    unsigned bits = (unsigned)__ballot(pred);  // wave32: low 32 bits
    cnt += pred ? 1u : 0u;
    if (lane == 0) {
      mask16[(size_t)(m0 + v) * MASK_LD + tn] = (unsigned short)(bits & 0xFFFFu);
      mask16[(size_t)(m0 + v + 8) * MASK_LD + tn] = (unsigned short)(bits >> 16);
    }
  }
  cnt += (unsigned)__shfl_xor((int)cnt, 16, 32);  // combine lane l / l+16 (same column)
  if (lane < 16 && cnt) atomicAdd(&deg[c], cnt);
}

__global__ __launch_bounds__(256) void k_dinv(const unsigned* __restrict__ deg,
                                              float* __restrict__ dinv) {
  int i = blockIdx.x * 256 + threadIdx.x;
  if (i < N_NODES) dinv[i] = rsqrtf((float)(deg[i] + 1u));  // +1 self-loop
}

// ------------------------------------- Y = dinv(row) * (Xin @ W), via WMMA
__global__ __launch_bounds__(256) void k_gemm_scale(
    const float* __restrict__ Xin, int ldx, int K, const float* __restrict__ W,
    const float* __restrict__ dinv, float* __restrict__ Y) {
  int lane = threadIdx.x & 31, wid = threadIdx.x >> 5;
  int tile = blockIdx.x * 8 + wid;            // 512*16 tiles
  int tm = tile >> 4, th = tile & 15;
  int q = lane & 15, half = lane >> 4, ka = half * 2;
  int m0 = tm * 16, h0 = th * 16;
  const float* ar = Xin + (size_t)(m0 + q) * ldx + ka;
  const float* bc = W + (size_t)ka * NHID + h0 + q;  // W[k][h], row-major
  v8f acc = {0.f, 0.f, 0.f, 0.f, 0.f, 0.f, 0.f, 0.f};
  for (int k = 0; k < K; k += 4) {
    v2f a = {ar[k], ar[k + 1]};
    const float* bp = bc + (size_t)k * NHID;
    v2f b = {bp[0], bp[NHID]};
    acc = wmma_f32(a, b, acc);
  }
  int mb = m0 + half * 8;
  for (int v = 0; v < 8; ++v) {
    int r = mb + v;
    Y[(size_t)r * NHID + h0 + q] = acc[v] * dinv[r];
  }
}

// ---- H[c] = relu( dinv[c]*(Y[c] + sum_{r<c,edge} Y[r]) + b ), fused pooling
// A-tile (0/1 in fp32) expanded from mask bits in registers; triangular K loop.
__global__ __launch_bounds__(256) void k_agg(
    const float* __restrict__ Y, const unsigned short* __restrict__ mask16,
    const float* __restrict__ dinv, const float* __restrict__ bias,
    float* __restrict__ Hout, float* __restrict__ maxp, float* __restrict__ sump) {
  int lane = threadIdx.x & 31, wid = threadIdx.x >> 5;
  int tile = blockIdx.x * 8 + wid;
  int tc = tile >> 4, th = tile & 15;
  int q = lane & 15, half = lane >> 4, ka = half * 2;
  int c0 = tc * 16, h0 = th * 16;
  v8f acc = {0.f, 0.f, 0.f, 0.f, 0.f, 0.f, 0.f, 0.f};
  const float* yb = Y + (size_t)ka * NHID + h0 + q;
  const unsigned short* mb = mask16 + (size_t)ka * MASK_LD + tc;
  int rmax = c0 + 16;                         // edges only for r < c
  for (int r0 = 0; r0 < rmax; r0 += 4) {
    unsigned w0 = mb[(size_t)r0 * MASK_LD];          // row r0+ka,   word tc
    unsigned w1 = mb[(size_t)(r0 + 1) * MASK_LD];    // row r0+ka+1, word tc
    v2f a = {(float)((w0 >> q) & 1u), (float)((w1 >> q) & 1u)};
    const float* yp = yb + (size_t)r0 * NHID;
    v2f b = {yp[0], yp[NHID]};
    acc = wmma_f32(a, b, acc);
  }
  int cb = c0 + half * 8;
  int h = h0 + q;
  float bh = bias[h];
  float mx = 0.f, sm = 0.f;
  for (int v = 0; v < 8; ++v) {
    int c = cb + v;
    float val = dinv[c] * (acc[v] + Y[(size_t)c * NHID + h]) + bh;  // +self-loop
    val = fmaxf(val, 0.f);                                          // relu
    Hout[(size_t)c * NHID + h] = val;
    mx = fmaxf(mx, val);
    sm += val;
  }
  mx = fmaxf(mx, __shfl_xor(mx, 16, 32));
  sm += __shfl_xor(sm, 16, 32);
  if (lane < 16) {                 // one lane per column h
    atomicMax((unsigned*)(maxp + h), __float_as_uint(mx));  // vals >= 0
    atomicAdd(sump + h, sm);
  }
}

__global__ __launch_bounds__(256) void k_final(const float* __restrict__ maxp,
                                               const float* __restrict__ sump,
                                               float* __restrict__ out) {
  int h = threadIdx.x;
  out[h] = maxp[h] + maxp[NHID + h] + maxp[2 * NHID + h];
  out[NHID + h] =
      (sump[h] + sump[NHID + h] + sump[2 * NHID + h]) * (1.f / (float)N_NODES);
}

// -------------------------------------------------------------------- launch
extern "C" void kernel_launch(void* const* d_in, const int* in_sizes, int n_in,
                              void* d_out, int out_size, void* d_ws,
                              size_t ws_size, hipStream_t stream) {
  const float* feat = (const float*)d_in[0];
  const float* W1 = (const float*)d_in[1];
  const float* b1 = (const float*)d_in[2];
  const float* W2 = (const float*)d_in[3];
  const float* b2 = (const float*)d_in[4];
  const float* W3 = (const float*)d_in[5];
  const float* b3 = (const float*)d_in[6];
  (void)in_sizes; (void)n_in; (void)out_size; (void)ws_size;

  char* ws = (char*)d_ws;
  size_t o = 0;
  // zero-init region (deg + pools), contiguous:
  unsigned* deg = (unsigned*)(ws + o);      o += (size_t)N_NODES * 4;
  float* maxp   = (float*)(ws + o);         o += 3 * NHID * 4;
  float* sump   = (float*)(ws + o);         o += 3 * NHID * 4;
  const int nzero = N_NODES + 6 * NHID;     // u32 words to zero
  // always-overwritten scratch:
  float* tmax   = (float*)(ws + o);         o += 256;                 // padded
  float* blkmax = (float*)(ws + o);         o += (size_t)64 * NT * 4; // 128 KB
  float* sq     = (float*)(ws + o);         o += (size_t)N_NODES * 4;
  float* dinv   = (float*)(ws + o);         o += (size_t)N_NODES * 4;
  unsigned short* mask16 = (unsigned short*)(ws + o);
  o += (size_t)N_NODES * MASK_LD * 2;                                  // 8 MB
  float* Ybuf = (float*)(ws + o);           o += (size_t)N_NODES * NHID * 4;
  float* H0   = (float*)(ws + o);           o += (size_t)N_NODES * NHID * 4;
  float* H1   = (float*)(ws + o);           o += (size_t)N_NODES * NHID * 4;
  // total ~= 34 MB

  k_zero<<<(nzero + 255) / 256, 256, 0, stream>>>(deg, nzero);
  k_sq<<<N_NODES / 8, 256, 0, stream>>>(feat, sq);
  k_gram_max<<<dim3(NT / 8, NT), 256, 0, stream>>>(feat, sq, blkmax);
  k_maxred<<<1, 256, 0, stream>>>(blkmax, 64 * NT, tmax);
  k_gram_mask<<<dim3(NT / 8, NT), 256, 0, stream>>>(feat, sq, tmax, mask16, deg);
  k_dinv<<<N_NODES / 256, 256, 0, stream>>>(deg, dinv);

  const int ntiles = NT * (NHID / 16);  // 8192 output tiles per stage
  // layer 1
  k_gemm_scale<<<ntiles / 8, 256, 0, stream>>>(feat, NFEAT, NFEAT, W1, dinv, Ybuf);
  k_agg<<<ntiles / 8, 256, 0, stream>>>(Ybuf, mask16, dinv, b1, H0, maxp, sump);
  // layer 2
  k_gemm_scale<<<ntiles / 8, 256, 0, stream>>>(H0, NHID, NHID, W2, dinv, Ybuf);
  k_agg<<<ntiles / 8, 256, 0, stream>>>(Ybuf, mask16, dinv, b2, H1, maxp + NHID,
                                        sump + NHID);
  // layer 3 (output activations H0 reused; only pools matter)
  k_gemm_scale<<<ntiles / 8, 256, 0, stream>>>(H1, NHID, NHID, W3, dinv, Ybuf);
  k_agg<<<ntiles / 8, 256, 0, stream>>>(Ybuf, mask16, dinv, b3, H0,
                                        maxp + 2 * NHID, sump + 2 * NHID);

  k_final<<<1, 256, 0, stream>>>(maxp, sump, (float*)d_out);
}